// GraphGCN_5471788335200
// MI455X (gfx1250) — compile-verified
//
#include <hip/hip_runtime.h>
#include <hip/hip_bf16.h>

typedef __attribute__((ext_vector_type(2))) float v2f;
typedef __attribute__((ext_vector_type(8))) float v8f;

#define FD 128     // feature dim (F_IN == H == 128)
#define NG 128     // graphs
#define NC 16      // classes

// ---------------------------------------------------------------- utilities
__global__ void fill_kernel(float* __restrict__ p, long long n, float v) {
    long long i = (long long)blockIdx.x * blockDim.x + threadIdx.x;
    long long stride = (long long)gridDim.x * blockDim.x;
    for (; i < n; i += stride) p[i] = v;
}

// deg[dst] += ew  (self-loop +1 folded into dinv_kernel)
__global__ void deg_kernel(const long long* __restrict__ ei,
                           const float* __restrict__ ew,
                           float* __restrict__ deg, int E) {
    int e = blockIdx.x * blockDim.x + threadIdx.x;
    if (e < E) unsafeAtomicAdd(&deg[ei[(long long)E + e]], ew[e]);
}

__global__ void dinv_kernel(float* __restrict__ deg, int n) {
    int i = blockIdx.x * blockDim.x + threadIdx.x;
    if (i < n) deg[i] = rsqrtf(deg[i] + 1.0f);
}

// ---------------------------------------------------------------- WMMA GEMM
// C[M x 128] = A[M x 128] @ W[128 x 128], fp32, V_WMMA_F32_16X16X4_F32.
// One block per 64-row super-tile (4 M-tiles); wave w owns column tile 16*w
// and keeps 4 accumulators, reusing each B fragment 4x.
#define LDSTR 132   // pad 128 -> 132 floats: conflict-free for both lane halves
__global__ __launch_bounds__(256) void gemm128_wmma(const float* __restrict__ A,
                                                    const float* __restrict__ W,
                                                    float* __restrict__ C, int M) {
    __shared__ float lds[128 * LDSTR + 64 * LDSTR];
    float* ldsW = lds;
    float* ldsA = lds + 128 * LDSTR;

    const int t  = threadIdx.x;
    const int m0 = blockIdx.x * 64;

    // stage W (128x128) into LDS, coalesced
#pragma unroll
    for (int i = 0; i < 64; ++i) {
        int idx = t + i * 256;
        ldsW[(idx >> 7) * LDSTR + (idx & 127)] = W[idx];
    }
    // stage A super-tile (64x128) into LDS, coalesced, zero-pad OOB rows
#pragma unroll
    for (int i = 0; i < 32; ++i) {
        int idx = t + i * 256;
        int r = idx >> 7, c = idx & 127;
        int row = m0 + r;
        ldsA[r * LDSTR + c] = (row < M) ? A[(long long)row * FD + c] : 0.0f;
    }
    __syncthreads();

    const int wave = t >> 5;
    const int lane = t & 31;
    const int half = lane >> 4;   // 0: K+0/K+1 , 1: K+2/K+3 (ISA 16x4 f32 layout)
    const int lr   = lane & 15;
    const int n0   = wave * 16;

    v8f acc0 = {}, acc1 = {}, acc2 = {}, acc3 = {};
#pragma unroll
    for (int k0 = 0; k0 < 128; k0 += 4) {
        const int kk = k0 + 2 * half;
        v2f b;
        b[0] = ldsW[kk * LDSTR + n0 + lr];
        b[1] = ldsW[(kk + 1) * LDSTR + n0 + lr];
        v2f a0, a1, a2, a3;
        a0[0] = ldsA[lr * LDSTR + kk];        a0[1] = ldsA[lr * LDSTR + kk + 1];
        a1[0] = ldsA[(16 + lr) * LDSTR + kk]; a1[1] = ldsA[(16 + lr) * LDSTR + kk + 1];
        a2[0] = ldsA[(32 + lr) * LDSTR + kk]; a2[1] = ldsA[(32 + lr) * LDSTR + kk + 1];
        a3[0] = ldsA[(48 + lr) * LDSTR + kk]; a3[1] = ldsA[(48 + lr) * LDSTR + kk + 1];
        acc0 = __builtin_amdgcn_wmma_f32_16x16x4_f32(false, a0, false, b, (short)0, acc0, false, false);
        acc1 = __builtin_amdgcn_wmma_f32_16x16x4_f32(false, a1, false, b, (short)0, acc1, false, false);
        acc2 = __builtin_amdgcn_wmma_f32_16x16x4_f32(false, a2, false, b, (short)0, acc2, false, false);
        acc3 = __builtin_amdgcn_wmma_f32_16x16x4_f32(false, a3, false, b, (short)0, acc3, false, false);
    }
    // D layout: VGPR r -> row (r + 8*half), col (n0 + lr)
#pragma unroll
    for (int r = 0; r < 8; ++r) {
        int row = m0 + r + 8 * half;
        if (row < M)      C[(long long)row * FD + n0 + lr]        = acc0[r];
        if (row + 16 < M) C[(long long)(row + 16) * FD + n0 + lr] = acc1[r];
        if (row + 32 < M) C[(long long)(row + 32) * FD + n0 + lr] = acc2[r];
        if (row + 48 < M) C[(long long)(row + 48) * FD + n0 + lr] = acc3[r];
    }
}

// ------------------------------------------------- self-loop + bias seeding
// agg[n][h] = xw[n][h] * dinv[n]^2 + bias[h]
__global__ void selfloop_bias_kernel(const float* __restrict__ xw,
                                     const float* __restrict__ dinv,
                                     const float* __restrict__ bias,
                                     float* __restrict__ agg, long long total) {
    long long i = (long long)blockIdx.x * blockDim.x + threadIdx.x;
    if (i >= total) return;
    int h = (int)(i & (FD - 1));
    long long n = i >> 7;
    float di = dinv[n];
    agg[i] = xw[i] * di * di + bias[h];
}

// ------------------------------------------------------------- edge scatter
// one wave per edge; lane carries 4 features; HW fp32 atomics into agg[dst]
__global__ __launch_bounds__(256) void edge_scatter_kernel(
        const long long* __restrict__ ei, const float* __restrict__ ew,
        const float* __restrict__ dinv, const float* __restrict__ xw,
        float* __restrict__ agg, int E) {
    int e = blockIdx.x * 8 + (threadIdx.x >> 5);
    if (e >= E) return;
    int lane = threadIdx.x & 31;
    long long s = ei[e];
    long long d = ei[(long long)E + e];
    float coeff = dinv[s] * ew[e] * dinv[d];
    float4 v = ((const float4*)(xw + s * FD))[lane];
    float* ad = agg + d * FD + lane * 4;
    unsafeAtomicAdd(ad + 0, v.x * coeff);
    unsafeAtomicAdd(ad + 1, v.y * coeff);
    unsafeAtomicAdd(ad + 2, v.z * coeff);
    unsafeAtomicAdd(ad + 3, v.w * coeff);
}

__global__ void relu_kernel(float* __restrict__ p, long long n) {
    long long i = (long long)blockIdx.x * blockDim.x + threadIdx.x;
    if (i < n) p[i] = fmaxf(p[i], 0.0f);
}

// ------------------------------------------------------------------ pooling
// h >= 0 after relu, so float max == uint max on the bit pattern (init 0)
__global__ void pool_kernel(const float* __restrict__ h,
                            const long long* __restrict__ batch,
                            float* __restrict__ mx, float* __restrict__ sum,
                            float* __restrict__ cnt, long long total) {
    long long i = (long long)blockIdx.x * blockDim.x + threadIdx.x;
    if (i >= total) return;
    long long n = i >> 7;
    int hh = (int)(i & (FD - 1));
    long long g = batch[n];
    float v = h[i];
    atomicMax((unsigned int*)&mx[g * FD + hh], __float_as_uint(v));
    unsafeAtomicAdd(&sum[g * FD + hh], v);
    if (hh == 0) unsafeAtomicAdd(&cnt[g], 1.0f);
}

// -------------------------------------------- final linear + log_softmax
__global__ void final_kernel(const float* __restrict__ mx,
                             const float* __restrict__ sum,
                             const float* __restrict__ cnt,
                             const float* __restrict__ lw,   // [2H, C]
                             const float* __restrict__ lb,   // [C]
                             float* __restrict__ out) {
    int g = threadIdx.x;
    if (g >= NG) return;
    float z[NC];
#pragma unroll
    for (int c = 0; c < NC; ++c) z[c] = lb[c];
    float invc = 1.0f / fmaxf(cnt[g], 1.0f);
    for (int k = 0; k < FD; ++k) {
        float a = mx[g * FD + k];
        float m = sum[g * FD + k] * invc;
#pragma unroll
        for (int c = 0; c < NC; ++c)
            z[c] += a * lw[k * NC + c] + m * lw[(FD + k) * NC + c];
    }
    float mz = z[0];
#pragma unroll
    for (int c = 1; c < NC; ++c) mz = fmaxf(mz, z[c]);
    float s = 0.0f;
#pragma unroll
    for (int c = 0; c < NC; ++c) s += expf(z[c] - mz);
    float ls = logf(s);
#pragma unroll
    for (int c = 0; c < NC; ++c) out[g * NC + c] = z[c] - mz - ls;
}

// ---------------------------------------------------------------- launcher
extern "C" void kernel_launch(void* const* d_in, const int* in_sizes, int n_in,
                              void* d_out, int out_size, void* d_ws, size_t ws_size,
                              hipStream_t stream) {
    const float*     x     = (const float*)d_in[0];
    const long long* ei    = (const long long*)d_in[1];   // int64 [2,E]
    const float*     ew    = (const float*)d_in[2];
    const long long* batch = (const long long*)d_in[3];   // int64 [N]
    const float*     W1    = (const float*)d_in[4];
    const float*     b1    = (const float*)d_in[5];
    const float*     W2    = (const float*)d_in[6];
    const float*     b2    = (const float*)d_in[7];
    const float*     lw    = (const float*)d_in[8];
    const float*     lb    = (const float*)d_in[9];
    float*           out   = (float*)d_out;

    const int N = in_sizes[0] / FD;     // 50000
    const int E = in_sizes[2];          // 600000
    const long long NH = (long long)N * FD;

    // workspace carve (floats, 64-float aligned chunks)
    float* p    = (float*)d_ws;
    float* dinv = p;  p += ((N + 63) / 64) * 64;
    float* xw   = p;  p += NH;                      // xw1, then xw2
    float* agg  = p;  p += NH;                      // agg1/h1, then agg2/h2
    float* mx   = p;  p += (long long)NG * FD;
    float* sum  = p;  p += (long long)NG * FD;
    float* cnt  = p;  p += NG;

    const int gGemm = (N + 63) / 64;                // 782 blocks, 64 rows each
    const int gNH   = (int)((NH + 255) / 256);
    const int gE    = (E + 255) / 256;
    const int gEdge = (E + 7) / 8;

    // degree -> dinv
    fill_kernel<<<256, 256, 0, stream>>>(dinv, N, 0.0f);
    deg_kernel<<<gE, 256, 0, stream>>>(ei, ew, dinv, E);
    dinv_kernel<<<(N + 255) / 256, 256, 0, stream>>>(dinv, N);

    // layer 1
    gemm128_wmma<<<gGemm, 256, 0, stream>>>(x, W1, xw, N);
    selfloop_bias_kernel<<<gNH, 256, 0, stream>>>(xw, dinv, b1, agg, NH);
    edge_scatter_kernel<<<gEdge, 256, 0, stream>>>(ei, ew, dinv, xw, agg, E);
    relu_kernel<<<gNH, 256, 0, stream>>>(agg, NH);        // agg -> h1

    // layer 2 (reuse buffers: xw <- h1 @ W2, then agg <- layer2 aggregate)
    gemm128_wmma<<<gGemm, 256, 0, stream>>>(agg, W2, xw, N);
    selfloop_bias_kernel<<<gNH, 256, 0, stream>>>(xw, dinv, b2, agg, NH);
    edge_scatter_kernel<<<gEdge, 256, 0, stream>>>(ei, ew, dinv, xw, agg, E);
    relu_kernel<<<gNH, 256, 0, stream>>>(agg, NH);        // agg -> h2

    // pooling (mx, sum, cnt are contiguous in the carve)
    fill_kernel<<<64, 256, 0, stream>>>(mx, (long long)NG * FD * 2 + NG, 0.0f);
    pool_kernel<<<gNH, 256, 0, stream>>>(agg, batch, mx, sum, cnt, NH);

    // classifier head
    final_kernel<<<1, NG, 0, stream>>>(mx, sum, cnt, lw, lb, out);
}